// SPGATNet_20005957665592
// MI455X (gfx1250) — compile-verified
//
#include <hip/hip_runtime.h>

#define N_NODES 100000
#define N_EDGES 1600000
#define NE_TOT  (N_EDGES + N_NODES)   // edges + self loops
#define C_IN    32
#define HEADS   4
#define C_OUT   32
#define HC      128                   // HEADS * C_OUT
#define NEG_SLOPE 0.2f

typedef __attribute__((ext_vector_type(2))) float v2f;
typedef __attribute__((ext_vector_type(8))) float v8f;

// ---------- sortable-uint encoding for float atomicMax ----------
__device__ __forceinline__ unsigned encf(float f) {
    unsigned b = __float_as_uint(f);
    return (b & 0x80000000u) ? ~b : (b | 0x80000000u);
}
__device__ __forceinline__ float decf(unsigned u) {
    return (u & 0x80000000u) ? __uint_as_float(u & 0x7FFFFFFFu)
                             : __uint_as_float(~u);
}

// ---------- 1) h = x @ W via V_WMMA_F32_16X16X4_F32 ----------
// Block = 256 threads = 8 waves. Wave w computes the 16x16 tile
// (mtile = blockIdx.x, ntile = w). K = 32 -> 8 chained WMMAs.
__global__ __launch_bounds__(256)
void gat_gemm_wmma(const float* __restrict__ x, const float* __restrict__ W,
                   float* __restrict__ h) {
    const int wave  = threadIdx.x >> 5;
    const int lane  = threadIdx.x & 31;
    const int mtile = blockIdx.x;
    const int ntile = wave;           // 8 ntiles of 16 cols = 128
    const int m     = lane & 15;      // M index within tile
    const int khalf = lane >> 4;      // 0 or 1 (K sub-half per ISA layout)
    const int row   = mtile * 16 + m;
    const float* __restrict__ xrow = x + (size_t)row * C_IN;
    const int ncol = ntile * 16 + m;  // B/N column this lane owns

    v8f acc = {};
    #pragma unroll
    for (int k = 0; k < 8; ++k) {
        const int kb = k * 4 + khalf * 2;
        v2f a, b;
        // A 16x4 f32 layout: VGPR r holds A[M=lane%16, K = k*4 + (lane/16)*2 + r]
        a.x = xrow[kb + 0];
        a.y = xrow[kb + 1];
        // B 4x16 f32, row-striped across lanes: VGPR r holds B[K = k*4 + (lane/16)*2 + r, N=lane%16]
        b.x = W[(size_t)(kb + 0) * HC + ncol];
        b.y = W[(size_t)(kb + 1) * HC + ncol];
        acc = __builtin_amdgcn_wmma_f32_16x16x4_f32(
                  false, a, false, b, (short)0, acc, false, false);
    }
    // D 16x16 f32: VGPR r -> h[M = r + khalf*8][N = lane%16]
    #pragma unroll
    for (int r = 0; r < 8; ++r) {
        h[(size_t)(mtile * 16 + r + khalf * 8) * HC + ncol] = acc[r];
    }
}

// ---------- 2) per-(node,head) attention logits ----------
__global__ __launch_bounds__(256)
void gat_alpha(const float* __restrict__ h, const float* __restrict__ a_src,
               const float* __restrict__ a_dst,
               float* __restrict__ asrc, float* __restrict__ adst) {
    const int idx = blockIdx.x * blockDim.x + threadIdx.x;  // n*HEADS + hd
    if (idx >= N_NODES * HEADS) return;
    const int n  = idx >> 2;
    const int hd = idx & 3;
    const float4* hv = (const float4*)(h + (size_t)n * HC + hd * C_OUT);
    const float4* sv = (const float4*)(a_src + hd * C_OUT);
    const float4* dv = (const float4*)(a_dst + hd * C_OUT);
    float ss = 0.f, sd = 0.f;
    #pragma unroll
    for (int i = 0; i < 8; ++i) {
        const float4 hh = hv[i], av = sv[i], bv = dv[i];
        ss += hh.x * av.x + hh.y * av.y + hh.z * av.z + hh.w * av.w;
        sd += hh.x * bv.x + hh.y * bv.y + hh.z * bv.z + hh.w * bv.w;
    }
    asrc[idx] = ss;
    adst[idx] = sd;
}

// ---------- 3) init m (enc(-inf) ordering min == 0), s, accum ----------
__global__ __launch_bounds__(256)
void gat_init(unsigned* __restrict__ mU, float* __restrict__ s,
              float* __restrict__ accum) {
    const int i = blockIdx.x * blockDim.x + threadIdx.x;
    if (i < N_NODES * HC) accum[i] = 0.f;
    if (i < N_NODES * HEADS) { mU[i] = 0u; s[i] = 0.f; }
}

// ---------- 4) edge pass 1: leaky-relu logits + segment max ----------
__global__ __launch_bounds__(256)
void edge_max(const long long* __restrict__ ei, const float* __restrict__ asrc,
              const float* __restrict__ adst, float* __restrict__ ebuf,
              unsigned* __restrict__ mU) {
    const int e = blockIdx.x * blockDim.x + threadIdx.x;
    if (e >= NE_TOT) return;
    int s_, d_;
    if (e < N_EDGES) { s_ = (int)ei[e]; d_ = (int)ei[N_EDGES + e]; }
    else             { s_ = d_ = e - N_EDGES; }
    #pragma unroll
    for (int hd = 0; hd < HEADS; ++hd) {
        float v = asrc[s_ * HEADS + hd] + adst[d_ * HEADS + hd];
        v = (v > 0.f) ? v : NEG_SLOPE * v;
        ebuf[(size_t)e * HEADS + hd] = v;
        atomicMax(&mU[d_ * HEADS + hd], encf(v));
    }
}

// ---------- 5) edge pass 2: exp(e - m[dst]) + segment sum ----------
__global__ __launch_bounds__(256)
void edge_expsum(const long long* __restrict__ ei, const unsigned* __restrict__ mU,
                 float* __restrict__ ebuf, float* __restrict__ sArr) {
    const int e = blockIdx.x * blockDim.x + threadIdx.x;
    if (e >= NE_TOT) return;
    int d_;
    if (e < N_EDGES) d_ = (int)ei[N_EDGES + e];
    else             d_ = e - N_EDGES;
    #pragma unroll
    for (int hd = 0; hd < HEADS; ++hd) {
        const float m  = decf(mU[d_ * HEADS + hd]);
        const float ex = expf(ebuf[(size_t)e * HEADS + hd] - m);
        ebuf[(size_t)e * HEADS + hd] = ex;
        atomicAdd(&sArr[d_ * HEADS + hd], ex);
    }
}

// ---------- 6) edge pass 3: wave-per-edge weighted scatter ----------
// 32 lanes x float4 = full 512B coalesced h[src] row; lane's head = lane/8.
__global__ __launch_bounds__(256)
void edge_scatter(const long long* __restrict__ ei, const float* __restrict__ ebuf,
                  const float* __restrict__ sArr, const float* __restrict__ h,
                  float* __restrict__ accum) {
    const int wave = threadIdx.x >> 5;
    const int lane = threadIdx.x & 31;
    const long long e = (long long)blockIdx.x * 8 + wave;
    if (e >= NE_TOT) return;
    int s_, d_;
    if (e < N_EDGES) { s_ = (int)ei[e]; d_ = (int)ei[N_EDGES + e]; }
    else             { s_ = d_ = (int)(e - N_EDGES); }
    const int hd = lane >> 3;
    const float alpha = ebuf[e * HEADS + hd] /
                        (sArr[d_ * HEADS + hd] + 1e-16f);
    const float4 hv = *(const float4*)(h + (size_t)s_ * HC + lane * 4);
    float* ap = accum + (size_t)d_ * HC + lane * 4;
    atomicAdd(ap + 0, hv.x * alpha);
    atomicAdd(ap + 1, hv.y * alpha);
    atomicAdd(ap + 2, hv.z * alpha);
    atomicAdd(ap + 3, hv.w * alpha);
}

// ---------- 7) mean over heads + bias + ELU ----------
__global__ __launch_bounds__(256)
void gat_finalize(const float* __restrict__ accum, const float* __restrict__ bias,
                  float* __restrict__ xout) {
    const int i = blockIdx.x * blockDim.x + threadIdx.x;  // n*32 + c
    if (i >= N_NODES * C_OUT) return;
    const int n = i >> 5, c = i & 31;
    const float* a = accum + (size_t)n * HC;
    float v = 0.25f * (a[c] + a[C_OUT + c] + a[2 * C_OUT + c] + a[3 * C_OUT + c])
              + bias[c];
    xout[i] = (v > 0.f) ? v : (expf(v) - 1.f);
}

extern "C" void kernel_launch(void* const* d_in, const int* in_sizes, int n_in,
                              void* d_out, int out_size, void* d_ws, size_t ws_size,
                              hipStream_t stream) {
    const float*     x0      = (const float*)d_in[0];
    const long long* ei      = (const long long*)d_in[1];   // int64 [2,E]
    const float*     Ws      = (const float*)d_in[3];       // [3,32,128]
    const float*     att_src = (const float*)d_in[4];       // [3,4,32]
    const float*     att_dst = (const float*)d_in[5];       // [3,4,32]
    const float*     bias    = (const float*)d_in[6];       // [3,32]
    float* out = (float*)d_out;

    // workspace layout (floats)
    float* ws    = (float*)d_ws;
    float* h     = ws;                                    // N*128
    float* accum = h     + (size_t)N_NODES * HC;          // N*128
    float* ebuf  = accum + (size_t)N_NODES * HC;          // NE_TOT*4
    float* asrc  = ebuf  + (size_t)NE_TOT * HEADS;        // N*4
    float* adst  = asrc  + (size_t)N_NODES * HEADS;       // N*4
    unsigned* mU = (unsigned*)(adst + (size_t)N_NODES * HEADS); // N*4
    float* sArr  = (float*)(mU + (size_t)N_NODES * HEADS);      // N*4
    float* xA    = sArr  + (size_t)N_NODES * HEADS;       // N*32
    float* xB    = xA    + (size_t)N_NODES * C_IN;        // N*32

    const float* xin = x0;
    for (int l = 0; l < 3; ++l) {
        float* xout = (l == 2) ? out : ((l == 0) ? xA : xB);
        const float* W  = Ws      + (size_t)l * C_IN * HC;
        const float* as = att_src + (size_t)l * HEADS * C_OUT;
        const float* ad = att_dst + (size_t)l * HEADS * C_OUT;
        const float* b  = bias    + (size_t)l * C_OUT;

        gat_gemm_wmma<<<N_NODES / 16, 256, 0, stream>>>(xin, W, h);
        gat_alpha<<<(N_NODES * HEADS + 255) / 256, 256, 0, stream>>>(h, as, ad, asrc, adst);
        gat_init<<<(N_NODES * HC + 255) / 256, 256, 0, stream>>>(mU, sArr, accum);
        edge_max<<<(NE_TOT + 255) / 256, 256, 0, stream>>>(ei, asrc, adst, ebuf, mU);
        edge_expsum<<<(NE_TOT + 255) / 256, 256, 0, stream>>>(ei, mU, ebuf, sArr);
        edge_scatter<<<(NE_TOT + 7) / 8, 256, 0, stream>>>(ei, ebuf, sArr, h, accum);
        gat_finalize<<<(N_NODES * C_OUT + 255) / 256, 256, 0, stream>>>(accum, b, xout);
        xin = xout;
    }
}